// GCNnet_80058190397973
// MI455X (gfx1250) — compile-verified
//
#include <hip/hip_runtime.h>

// ---------------- problem constants ----------------
#define NN   20000       // nodes
#define EE   160000      // edges
#define NB   64          // graphs in batch
#define NHD  8           // GAT heads
#define CPH  32          // channels per head
#define F0   32
#define D1   256         // H*C
#define D2   512
#define D3   1024
#define TOT_E (EE + NN)  // edges + self loops

// ---------------- workspace layout (float offsets) ----------------
#define OFF_H      0u          // [NN*D1]   GAT hidden h
#define OFF_SCS    5120000u    // [NN*NHD]  src scores
#define OFF_SCD    5280000u    // [NN*NHD]  dst scores
#define OFF_MKEY   5440000u    // [NN*NHD]  segment-max keys (uint)
#define OFF_ESUM   5600000u    // [NN*NHD]  softmax denominators
#define OFF_EEXP   5760000u    // [TOT_E*NHD] exp(e - max)
#define OFF_X1     7200000u    // [NN*D1]   GAT output
#define OFF_HH     12320000u   // [NN*D2]   x1 @ W2
#define OFF_HH2    0u          // [NN*D3]   x2 @ W3 (aliases dead GAT region)
#define OFF_X2     22560000u   // [NN*D2]
#define OFF_X3     32800000u   // [NN*D3]
#define OFF_DEG    53280000u   // [NN]
#define OFF_SNORM  53300000u   // [NN]
#define OFF_NORM   53320000u   // [EE]
#define OFF_GMAX   53480000u   // [NB*D3]
#define OFF_GSUM   53545536u   // [NB*D3]
#define OFF_CNT    53611072u   // [NB]
#define OFF_P      53611136u   // [NB*2048]
#define OFF_H1     53742208u   // [NB*256]

typedef __attribute__((ext_vector_type(2))) float v2f;
typedef __attribute__((ext_vector_type(8))) float v8f;

__device__ __forceinline__ unsigned fkey(float f) {
  unsigned u = __float_as_uint(f);
  return (u & 0x80000000u) ? ~u : (u | 0x80000000u);
}
__device__ __forceinline__ float unkey(unsigned k) {
  return __uint_as_float((k & 0x80000000u) ? (k & 0x7fffffffu) : ~k);
}

#define WMMA_F32(A, B, C) \
  __builtin_amdgcn_wmma_f32_16x16x4_f32(false, (A), false, (B), (short)0, (C), false, false)

// ---------------- generic fill ----------------
__global__ void fill_f32(float* __restrict__ p, int n, float v) {
  int i = blockIdx.x * blockDim.x + threadIdx.x;
  if (i < n) p[i] = v;
}

// ---------------- fp32 WMMA GEMM: Y[Nn,Mm] = X[Nn,Kk] @ W[Kk,Mm] ----------------
// One wave computes a 16x64 output slab = 4 independent 16x16 WMMA tiles sharing the
// (strided, expensive) A fragment. K stepped by 4 via V_WMMA_F32_16X16X4_F32.
// A layout (16x4, ISA 7.12.2): lanes 0-15 hold M=0..15 K={0,1}; lanes 16-31 K={2,3}.
// B layout (4x16): lanes 0-15 hold N=0..15 K={0,1}; lanes 16-31 K={2,3}.
// C/D layout: vgpr r -> row r (lanes 0-15) / row r+8 (lanes 16-31), col = lane&15.
template <int Kk, int Mm>
__global__ void gemm_wmma_f32(const float* __restrict__ X, const float* __restrict__ W,
                              float* __restrict__ Y, int Nn) {
  constexpr int TILESM = Mm / 64;
  int wave = (int)((blockIdx.x * blockDim.x + threadIdx.x) >> 5);
  int lane = threadIdx.x & 31;
  int tiles = (Nn >> 4) * TILESM;
  if (wave >= tiles) return;               // wave-uniform exit; EXEC stays all-1s for WMMA
  int tm = wave % TILESM, tn = wave / TILESM;
  int row0 = tn << 4, col0 = tm << 6;
  int lid  = lane & 15, half = lane >> 4;

  const float* Arow = X + (size_t)(row0 + lid) * Kk + 2 * half;
  const float* Bp   = W + (size_t)(2 * half) * Mm + col0 + lid;

  v8f acc0 = {0.f, 0.f, 0.f, 0.f, 0.f, 0.f, 0.f, 0.f};
  v8f acc1 = acc0, acc2 = acc0, acc3 = acc0;
#pragma unroll 4
  for (int k = 0; k < Kk; k += 4) {
    v2f a = *(const v2f*)(Arow + k);              // 8B aligned: row*Kk + 2*half + k is even
    const float* b = Bp + (size_t)k * Mm;
    v2f b0; b0.x = b[0];  b0.y = b[Mm];
    v2f b1; b1.x = b[16]; b1.y = b[16 + Mm];
    v2f b2; b2.x = b[32]; b2.y = b[32 + Mm];
    v2f b3; b3.x = b[48]; b3.y = b[48 + Mm];
    acc0 = WMMA_F32(a, b0, acc0);
    acc1 = WMMA_F32(a, b1, acc1);
    acc2 = WMMA_F32(a, b2, acc2);
    acc3 = WMMA_F32(a, b3, acc3);
  }
  float* Yp = Y + (size_t)(row0 + 8 * half) * Mm + col0 + lid;
#pragma unroll
  for (int r = 0; r < 8; ++r) {
    Yp[(size_t)r * Mm + 0]  = acc0[r];
    Yp[(size_t)r * Mm + 16] = acc1[r];
    Yp[(size_t)r * Mm + 32] = acc2[r];
    Yp[(size_t)r * Mm + 48] = acc3[r];
  }
}

// ---------------- GAT attention scores ----------------
__global__ void gat_scores(const float* __restrict__ h, const float* __restrict__ a_src,
                           const float* __restrict__ a_dst, float* __restrict__ sc_s,
                           float* __restrict__ sc_d) {
  int i = blockIdx.x * blockDim.x + threadIdx.x;   // n*NHD + hd
  if (i >= NN * NHD) return;
  int n = i >> 3, hd = i & 7;
  const float* hp = h + (size_t)n * D1 + hd * CPH;
  const float* as = a_src + hd * CPH;
  const float* ad = a_dst + hd * CPH;
  float ss = 0.f, sd = 0.f;
#pragma unroll 8
  for (int c = 0; c < CPH; ++c) { float v = hp[c]; ss += v * as[c]; sd += v * ad[c]; }
  sc_s[i] = ss; sc_d[i] = sd;
}

// ---------------- GAT softmax pass 1: segment max via monotone-key atomicMax ----------------
__global__ void gat_edge_max(const int* __restrict__ srcI, const int* __restrict__ dstI,
                             const float* __restrict__ sc_s, const float* __restrict__ sc_d,
                             unsigned* __restrict__ mkey) {
  int i = blockIdx.x * blockDim.x + threadIdx.x;
  if (i >= TOT_E * NHD) return;
  int e = i >> 3, hd = i & 7;
  int s, d;
  if (e < EE) { s = srcI[e]; d = dstI[e]; } else { s = d = e - EE; }
  float v = sc_s[s * NHD + hd] + sc_d[d * NHD + hd];
  v = v > 0.f ? v : 0.2f * v;                       // leaky_relu(0.2)
  atomicMax(&mkey[d * NHD + hd], fkey(v));
}

// ---------------- GAT softmax pass 2: exp + segment sum ----------------
__global__ void gat_edge_exp(const int* __restrict__ srcI, const int* __restrict__ dstI,
                             const float* __restrict__ sc_s, const float* __restrict__ sc_d,
                             const unsigned* __restrict__ mkey, float* __restrict__ eexp,
                             float* __restrict__ esum) {
  int i = blockIdx.x * blockDim.x + threadIdx.x;
  if (i >= TOT_E * NHD) return;
  int e = i >> 3, hd = i & 7;
  int s, d;
  if (e < EE) { s = srcI[e]; d = dstI[e]; } else { s = d = e - EE; }
  float v = sc_s[s * NHD + hd] + sc_d[d * NHD + hd];
  v = v > 0.f ? v : 0.2f * v;
  float ex = expf(v - unkey(mkey[d * NHD + hd]));
  eexp[i] = ex;
  atomicAdd(&esum[d * NHD + hd], ex);
}

// ---------------- GAT aggregation: x1[d] += alpha * h[s] ----------------
__global__ void gat_aggregate(const int* __restrict__ srcI, const int* __restrict__ dstI,
                              const float* __restrict__ eexp, const float* __restrict__ esum,
                              const float* __restrict__ h, float* __restrict__ x1) {
  int i = blockIdx.x * blockDim.x + threadIdx.x;
  if (i >= TOT_E * NHD) return;
  int e = i >> 3, hd = i & 7;
  int s, d;
  if (e < EE) { s = srcI[e]; d = dstI[e]; } else { s = d = e - EE; }
  float alpha = eexp[i] / esum[d * NHD + hd];
  const float* hp = h + (size_t)s * D1 + hd * CPH;
  float* op = x1 + (size_t)d * D1 + hd * CPH;
#pragma unroll 8
  for (int c = 0; c < CPH; ++c) atomicAdd(&op[c], alpha * hp[c]);
}

// ---------------- bias + relu (in place) ----------------
__global__ void bias_relu(float* __restrict__ p, const float* __restrict__ b, int n, int period) {
  int i = blockIdx.x * blockDim.x + threadIdx.x;
  if (i < n) p[i] = fmaxf(p[i] + b[i % period], 0.f);
}

// ---------------- GCN degree / norm ----------------
__global__ void deg_accum(const int* __restrict__ dstI, const float* __restrict__ ew,
                          float* __restrict__ deg) {
  int e = blockIdx.x * blockDim.x + threadIdx.x;
  if (e < EE) atomicAdd(&deg[dstI[e]], ew[e]);
}
__global__ void edge_norm(const int* __restrict__ srcI, const int* __restrict__ dstI,
                          const float* __restrict__ ew, const float* __restrict__ deg,
                          float* __restrict__ nrm) {
  int e = blockIdx.x * blockDim.x + threadIdx.x;
  if (e < EE) nrm[e] = rsqrtf(deg[srcI[e]]) * ew[e] * rsqrtf(deg[dstI[e]]);
}
__global__ void node_snorm(const float* __restrict__ deg, float* __restrict__ snorm) {
  int n = blockIdx.x * blockDim.x + threadIdx.x;
  if (n < NN) { float di = rsqrtf(deg[n]); snorm[n] = di * di; }
}

// ---------------- GCN edge aggregation: acc[dst] += norm[e] * hh[src] ----------------
// grid.x = EE, blockDim = M/4 (float4 per thread)
__global__ void gcn_agg(const int* __restrict__ srcI, const int* __restrict__ dstI,
                        const float* __restrict__ nrm, const float* __restrict__ hh,
                        float* __restrict__ acc, int M) {
  int e = blockIdx.x;
  int f4 = threadIdx.x;
  int s = srcI[e], d = dstI[e];
  float nv = nrm[e];
  float4 v = ((const float4*)(hh + (size_t)s * M))[f4];
  float* op = acc + (size_t)d * M + f4 * 4;
  atomicAdd(op + 0, nv * v.x);
  atomicAdd(op + 1, nv * v.y);
  atomicAdd(op + 2, nv * v.z);
  atomicAdd(op + 3, nv * v.w);
}

// ---------------- GCN epilogue: + self term + bias, BN(eval), ReLU (in place on acc) ----------------
__global__ void gcn_finalize(float* __restrict__ acc, const float* __restrict__ hh,
                             const float* __restrict__ snorm, const float* __restrict__ bias,
                             const float* __restrict__ g, const float* __restrict__ be,
                             const float* __restrict__ rm, const float* __restrict__ rv,
                             int mshift) {
  int i = blockIdx.x * blockDim.x + threadIdx.x;
  int M = 1 << mshift;
  if (i >= NN * M) return;
  int n = i >> mshift, f = i & (M - 1);
  float v = acc[i] + snorm[n] * hh[i] + bias[f];
  v = (v - rm[f]) * rsqrtf(rv[f] + 1e-5f) * g[f] + be[f];
  acc[i] = fmaxf(v, 0.f);
}

// ---------------- pooling ----------------
__global__ void pool_maxsum(const float* __restrict__ x3, const int* __restrict__ batch,
                            float* __restrict__ gmax, float* __restrict__ gsum) {
  int i = blockIdx.x * blockDim.x + threadIdx.x;
  if (i >= NN * D3) return;
  int n = i >> 10, f = i & (D3 - 1);
  int b = batch[n];
  float v = x3[i];                                  // >= 0 after ReLU
  atomicMax((unsigned*)&gmax[b * D3 + f], __float_as_uint(v));
  atomicAdd(&gsum[b * D3 + f], v);
}
__global__ void count_nodes(const int* __restrict__ batch, float* __restrict__ cnt) {
  int n = blockIdx.x * blockDim.x + threadIdx.x;
  if (n < NN) atomicAdd(&cnt[batch[n]], 1.f);
}
__global__ void assemble_p(const float* __restrict__ gmax, const float* __restrict__ gsum,
                           const float* __restrict__ cnt, float* __restrict__ p) {
  int i = blockIdx.x * blockDim.x + threadIdx.x;
  if (i >= NB * D3) return;
  int b = i >> 10, f = i & (D3 - 1);
  p[b * 2048 + f]       = gmax[i];
  p[b * 2048 + D3 + f]  = gsum[i] / fmaxf(cnt[b], 1.f);
}

// ---------------- MLP head ----------------
__global__ void mlp1(const float* __restrict__ p, const float* __restrict__ Wf1,
                     const float* __restrict__ bf1, float* __restrict__ h1) {
  int i = blockIdx.x * blockDim.x + threadIdx.x;   // b*256 + j
  if (i >= NB * 256) return;
  int b = i >> 8, j = i & 255;
  const float* pp = p + b * 2048;
  float s = bf1[j];
  for (int k = 0; k < 2048; ++k) s += pp[k] * Wf1[k * 256 + j];
  h1[i] = fmaxf(s, 0.f);
}
__global__ void mlp2(const float* __restrict__ h1, const float* __restrict__ Wf2,
                     const float* __restrict__ bf2, float* __restrict__ out) {
  int i = blockIdx.x * blockDim.x + threadIdx.x;   // b*2 + k
  if (i >= NB * 2) return;
  int b = i >> 1, k2 = i & 1;
  float s = bf2[k2];
  for (int j = 0; j < 256; ++j) s += h1[b * 256 + j] * Wf2[j * 2 + k2];
  out[i] = s;
}

static inline int cdiv(long long n, int d) { return (int)((n + d - 1) / d); }

extern "C" void kernel_launch(void* const* d_in, const int* in_sizes, int n_in,
                              void* d_out, int out_size, void* d_ws, size_t ws_size,
                              hipStream_t stream) {
  (void)in_sizes; (void)n_in; (void)out_size; (void)ws_size;
  const float* x     = (const float*)d_in[0];
  const float* ew    = (const float*)d_in[1];
  const float* Wg    = (const float*)d_in[2];
  const float* a_src = (const float*)d_in[3];
  const float* a_dst = (const float*)d_in[4];
  const float* bg    = (const float*)d_in[5];
  const float* W2    = (const float*)d_in[6];
  const float* b2    = (const float*)d_in[7];
  const float* W3    = (const float*)d_in[8];
  const float* b3    = (const float*)d_in[9];
  const float* g1    = (const float*)d_in[10];
  const float* be1   = (const float*)d_in[11];
  const float* rm1   = (const float*)d_in[12];
  const float* rv1   = (const float*)d_in[13];
  const float* g2    = (const float*)d_in[14];
  const float* be2   = (const float*)d_in[15];
  const float* rm2   = (const float*)d_in[16];
  const float* rv2   = (const float*)d_in[17];
  const float* Wf1   = (const float*)d_in[18];
  const float* bf1   = (const float*)d_in[19];
  const float* Wf2   = (const float*)d_in[20];
  const float* bf2   = (const float*)d_in[21];
  const int*   ei    = (const int*)d_in[22];
  const int*   batch = (const int*)d_in[23];
  const int* srcI = ei;
  const int* dstI = ei + EE;

  float*    ws    = (float*)d_ws;
  float*    h     = ws + OFF_H;
  float*    sc_s  = ws + OFF_SCS;
  float*    sc_d  = ws + OFF_SCD;
  unsigned* mkey  = (unsigned*)(ws + OFF_MKEY);
  float*    esum  = ws + OFF_ESUM;
  float*    eexp  = ws + OFF_EEXP;
  float*    x1    = ws + OFF_X1;
  float*    hh    = ws + OFF_HH;
  float*    hh2   = ws + OFF_HH2;
  float*    x2    = ws + OFF_X2;
  float*    x3    = ws + OFF_X3;
  float*    deg   = ws + OFF_DEG;
  float*    snorm = ws + OFF_SNORM;
  float*    nrm   = ws + OFF_NORM;
  float*    gmax  = ws + OFF_GMAX;
  float*    gsum  = ws + OFF_GSUM;
  float*    cnt   = ws + OFF_CNT;
  float*    pbuf  = ws + OFF_P;
  float*    h1    = ws + OFF_H1;
  float*    out   = (float*)d_out;

  const int T = 256;

  // ---- init accumulators ----
  fill_f32<<<cdiv(NN * NHD, T), T, 0, stream>>>((float*)mkey, NN * NHD, 0.f);
  fill_f32<<<cdiv(NN * NHD, T), T, 0, stream>>>(esum, NN * NHD, 0.f);
  fill_f32<<<cdiv((long long)NN * D1, T), T, 0, stream>>>(x1, NN * D1, 0.f);
  fill_f32<<<cdiv(NN, T), T, 0, stream>>>(deg, NN, 1.f);

  // ---- GAT: h = x @ Wg  (N x 32 @ 32 x 256) ----
  gemm_wmma_f32<F0, D1><<<cdiv((NN / 16) * (D1 / 64), 8), T, 0, stream>>>(x, Wg, h, NN);
  gat_scores<<<cdiv(NN * NHD, T), T, 0, stream>>>(h, a_src, a_dst, sc_s, sc_d);
  gat_edge_max<<<cdiv((long long)TOT_E * NHD, T), T, 0, stream>>>(srcI, dstI, sc_s, sc_d, mkey);
  gat_edge_exp<<<cdiv((long long)TOT_E * NHD, T), T, 0, stream>>>(srcI, dstI, sc_s, sc_d, mkey, eexp, esum);
  gat_aggregate<<<cdiv((long long)TOT_E * NHD, T), T, 0, stream>>>(srcI, dstI, eexp, esum, h, x1);
  bias_relu<<<cdiv((long long)NN * D1, T), T, 0, stream>>>(x1, bg, NN * D1, D1);

  // ---- GCN normalization ----
  deg_accum<<<cdiv(EE, T), T, 0, stream>>>(dstI, ew, deg);
  edge_norm<<<cdiv(EE, T), T, 0, stream>>>(srcI, dstI, ew, deg, nrm);
  node_snorm<<<cdiv(NN, T), T, 0, stream>>>(deg, snorm);

  // ---- GCN layer 1: hh = x1 @ W2 ; x2 = BN(agg + self + b2) ReLU ----
  gemm_wmma_f32<D1, D2><<<cdiv((NN / 16) * (D2 / 64), 8), T, 0, stream>>>(x1, W2, hh, NN);
  fill_f32<<<cdiv((long long)NN * D2, T), T, 0, stream>>>(x2, NN * D2, 0.f);
  gcn_agg<<<EE, D2 / 4, 0, stream>>>(srcI, dstI, nrm, hh, x2, D2);
  gcn_finalize<<<cdiv((long long)NN * D2, T), T, 0, stream>>>(x2, hh, snorm, b2, g1, be1, rm1, rv1, 9);

  // ---- GCN layer 2: hh2 = x2 @ W3 ; x3 = BN(agg + self + b3) ReLU ----
  gemm_wmma_f32<D2, D3><<<cdiv((NN / 16) * (D3 / 64), 8), T, 0, stream>>>(x2, W3, hh2, NN);
  fill_f32<<<cdiv((long long)NN * D3, T), T, 0, stream>>>(x3, NN * D3, 0.f);
  gcn_agg<<<EE, D3 / 4, 0, stream>>>(srcI, dstI, nrm, hh2, x3, D3);
  gcn_finalize<<<cdiv((long long)NN * D3, T), T, 0, stream>>>(x3, hh2, snorm, b3, g2, be2, rm2, rv2, 10);

  // ---- global max + mean pool ----
  fill_f32<<<cdiv(NB * D3, T), T, 0, stream>>>(gmax, NB * D3, 0.f);
  fill_f32<<<cdiv(NB * D3, T), T, 0, stream>>>(gsum, NB * D3, 0.f);
  fill_f32<<<cdiv(NB, T), T, 0, stream>>>(cnt, NB, 0.f);
  pool_maxsum<<<cdiv((long long)NN * D3, T), T, 0, stream>>>(x3, batch, gmax, gsum);
  count_nodes<<<cdiv(NN, T), T, 0, stream>>>(batch, cnt);
  assemble_p<<<cdiv(NB * D3, T), T, 0, stream>>>(gmax, gsum, cnt, pbuf);

  // ---- MLP head ----
  mlp1<<<cdiv(NB * 256, T), T, 0, stream>>>(pbuf, Wf1, bf1, h1);
  mlp2<<<1, 128, 0, stream>>>(h1, Wf2, bf2, out);
}